// VoteFilterLoss_57930518889028
// MI455X (gfx1250) — compile-verified
//
#include <hip/hip_runtime.h>
#include <hip/hip_bf16.h>

// ---------------------------------------------------------------------------
// VoteFilterLoss for MI455X (gfx1250, wave32).
//   loss = [ sum_{t==0} -log1p(-p)  -  sum_{runs of ones} FSA_score(run) ] / N
// FSA (FILTER=3, MAJORITY=2) specialized to 5 live states (see analysis).
// Row split into chunks; each thread exactly owns the runs starting in its
// chunk (skip left-continuation, overshoot right until its run ends).
// Wave reduction via V_WMMA_F32_16X16X4_F32.
// ---------------------------------------------------------------------------

#define NEGV (-1.0e30f)

typedef __attribute__((ext_vector_type(2))) float v2f;
typedef __attribute__((ext_vector_type(8))) float v8f;

__device__ __forceinline__ float logadd(float x, float y) {
    float mx = fmaxf(x, y);
    return mx + log1pf(__expf(-fabsf(x - y)));
}

__global__ void __launch_bounds__(256)
vote_filter_loss_partials(const float* __restrict__ outp,
                          const float* __restrict__ tgtp,
                          float* __restrict__ partials,
                          int T, int chunks_per_row) {
    const int CHUNK = T / chunks_per_row;
    int gid  = blockIdx.x * blockDim.x + threadIdx.x;
    int row  = gid / chunks_per_row;
    int ck   = gid - row * chunks_per_row;
    long base = (long)row * (long)T;
    const float* po = outp + base;
    const float* pt = tgtp + base;
    int c0 = ck * CHUNK;
    int c1 = c0 + CHUNK;

    int t = c0;
    // A run of ones continuing in from the left belongs to the previous
    // chunk's owner: skip it entirely.
    if (c0 > 0 && pt[c0 - 1] > 0.5f) {
        while (t < T && pt[t] > 0.5f) ++t;
    }

    // Log-semiring forward state: a0 = "start" state (0 iff ready to begin a
    // segment), a1..a4 = live window states. msave caches last step's m so the
    // finalize is logadd(msave, a3, a4).
    float a0 = 0.0f, a1 = NEGV, a2 = NEGV, a3 = NEGV, a4 = NEGV;
    float msave = NEGV;
    float acc = 0.0f;
    bool  prev = false;

    // Main chunk scan: fully branch-free per element (v_cndmask selects),
    // since random targets make wave-level branches always divergent.
    for (; t < c1; ++t) {
        float p   = po[t];
        bool  one = pt[t] > 0.5f;
        float lp0 = log1pf(-p);     // log(1-p)
        float lp1 = __logf(p);      // log(p)
        float m   = logadd(logadd(a1, a4), a0);
        float fin = logadd(logadd(a3, a4), msave);
        float s1 = lp1 + m, s2 = lp0 + m, s3 = lp1 + a2, s4 = lp1 + a3;
        bool endseg = (!one) && prev;
        acc   = endseg ? (acc - fin) : acc;   // lattice score of finished run
        acc   = one ? acc : (acc - lp0);      // BCE at target==0 positions
        a0    = one ? NEGV : 0.0f;
        a1    = one ? s1 : NEGV;
        a2    = one ? s2 : NEGV;
        a3    = one ? s3 : NEGV;
        a4    = one ? s4 : NEGV;
        msave = one ? m  : NEGV;
        prev  = one;
    }

    // Overshoot: finish a run that crosses the right chunk boundary.
    while (prev && t < T) {
        float p   = po[t];
        bool  one = pt[t] > 0.5f;
        if (!one) {
            acc -= logadd(logadd(a3, a4), msave);
            prev = false;
        } else {
            float lp0 = log1pf(-p);
            float lp1 = __logf(p);
            float m   = logadd(logadd(a1, a4), a0);
            float s1 = lp1 + m, s2 = lp0 + m, s3 = lp1 + a2, s4 = lp1 + a3;
            a0 = NEGV; a1 = s1; a2 = s2; a3 = s3; a4 = s4; msave = m;
        }
        ++t;
    }
    if (prev) acc -= logadd(logadd(a3, a4), msave);   // run touching row end

    // ---- wave32 reduction via one V_WMMA_F32_16X16X4_F32 -------------------
    // A(16x4): lane L, VGPR0 -> A[L%16][2*(L/16)]; VGPR1 zeroed.
    // B(4x16) = all ones  =>  D[m][n] = acc_m + acc_{m+16} (broadcast over n).
    // Sum of 8 D VGPRs is uniform per half-wave; shfl_xor(16) finishes it.
    float wsum;
#if __has_builtin(__builtin_amdgcn_wmma_f32_16x16x4_f32)
    {
        v2f A; A[0] = acc;  A[1] = 0.0f;
        v2f Bm; Bm[0] = 1.0f; Bm[1] = 1.0f;
        v8f C = {};
        v8f D = __builtin_amdgcn_wmma_f32_16x16x4_f32(
            /*neg_a=*/false, A, /*neg_b=*/false, Bm,
            /*c_mod=*/(short)0, C, /*reuse_a=*/false, /*reuse_b=*/false);
        float s = ((D[0] + D[1]) + (D[2] + D[3])) + ((D[4] + D[5]) + (D[6] + D[7]));
        wsum = s + __shfl_xor(s, 16, 32);
    }
#else
    {
        float s = acc;
        for (int d = 16; d >= 1; d >>= 1) s += __shfl_xor(s, d, 32);
        wsum = s;
    }
#endif

    __shared__ float wave_part[8];   // 256 threads = 8 wave32s
    int lane = threadIdx.x & 31;
    int wid  = threadIdx.x >> 5;
    if (lane == 0) wave_part[wid] = wsum;
    __syncthreads();
    if (threadIdx.x == 0) {
        float b = 0.0f;
        int nw = blockDim.x >> 5;
        for (int i = 0; i < nw; ++i) b += wave_part[i];
        partials[blockIdx.x] = b;    // deterministic per-block partial
    }
}

__global__ void vote_filter_loss_final(const float* __restrict__ partials,
                                       int n, float* __restrict__ out,
                                       float invN) {
    if (blockIdx.x == 0 && threadIdx.x == 0) {
        float s = 0.0f;
        for (int i = 0; i < n; ++i) s += partials[i];  // fixed order: deterministic
        out[0] = s * invN;
    }
}

extern "C" void kernel_launch(void* const* d_in, const int* in_sizes, int n_in,
                              void* d_out, int out_size, void* d_ws, size_t ws_size,
                              hipStream_t stream) {
    (void)n_in; (void)out_size; (void)ws_size;
    const float* outp = (const float*)d_in[0];   // probabilities p
    const float* tgtp = (const float*)d_in[1];   // targets 0/1 (float32)

    const int T = 4096;
    long total = (long)in_sizes[0];
    int  B = (int)(total / T);

    const int CHUNKS_PER_ROW = 16;               // chunk = 256 elems
    long nthreads = (long)B * CHUNKS_PER_ROW;    // 65536 threads = 2048 waves
    int  block = 256;
    int  grid  = (int)((nthreads + block - 1) / block);

    float* partials = (float*)d_ws;
    vote_filter_loss_partials<<<grid, block, 0, stream>>>(outp, tgtp, partials,
                                                          T, CHUNKS_PER_ROW);
    float invN = (float)(1.0 / ((double)B * (double)T));
    vote_filter_loss_final<<<1, 32, 0, stream>>>(partials, grid,
                                                 (float*)d_out, invN);
}